// BorderAlign_14972255994019
// MI455X (gfx1250) — compile-verified
//
#include <hip/hip_runtime.h>
#include <stdint.h>

// Problem constants (from the reference): B=2, C4=128 -> C=32, H=W=128,
// K=H*W=16384 boxes, pool_size=10 -> P=11 samples per border.
#define Bn      2
#define Cn      32
#define Hn      128
#define Wn      128
#define Kn      (Hn * Wn)      // 16384
#define Pn      11
#define KTILE   2048
#define KT      (Kn / KTILE)   // 8 k-tiles
#define THREADS 256
#define NCH     2              // channel planes staged per block (2*64KB LDS)
#define KPT     (KTILE / THREADS)  // 8 boxes per thread

static_assert(KT == 8 && (Cn % NCH) == 0 && KPT == 8, "decode assumptions");

typedef __attribute__((ext_vector_type(4))) unsigned int u32x4;
typedef __attribute__((ext_vector_type(8))) int          i32x8;
typedef __attribute__((ext_vector_type(4))) int          i32x4;

// ---- CDNA5 Tensor Data Mover availability / arity detection ----------------
#if defined(__has_builtin)
#  if __has_builtin(__builtin_amdgcn_tensor_load_to_lds) && \
      __has_builtin(__builtin_amdgcn_s_wait_tensorcnt)
#    define HAVE_TDM 1
#  endif
#endif
#ifndef HAVE_TDM
#  define HAVE_TDM 0
#endif

#if HAVE_TDM
#  if defined(__has_include)
#    if __has_include(<hip/amd_detail/amd_gfx1250_TDM.h>)
#      define TDM_ARGS 6   // amdgpu-toolchain (clang-23 / therock headers)
#    else
#      define TDM_ARGS 5   // ROCm 7.2 clang-22
#    endif
#  else
#    define TDM_ARGS 5
#  endif
#endif

#if HAVE_TDM
// Issue one TDM copy of a 128x128 fp32 plane (64 KB) global -> LDS.
// Descriptor layout per CDNA5 ISA.
__device__ __forceinline__ void tdm_load_plane(const float* __restrict__ gsrc,
                                               float* lds_tile) {
  const uint64_t ga = (uint64_t)(uintptr_t)gsrc;
  // Generic LDS pointer: low 32 bits = byte offset in the workgroup's LDS.
  const uint32_t la = (uint32_t)(uintptr_t)lds_tile;

  u32x4 g0;
  g0.x = 1u;                                    // count=1, user descriptor
  g0.y = la;                                    // lds_addr (bytes)
  g0.z = (uint32_t)(ga & 0xFFFFFFFFu);          // global_addr[31:0]
  g0.w = (uint32_t)((ga >> 32) & 0x01FFFFFFu)   // global_addr[56:32]
       | (2u << 30);                            // type=2 ("image")

  i32x8 g1;
  g1[0] = 0x00020000;        // workgroup_mask=0, data_size=2 (4 bytes)
  g1[1] = (int)(Wn << 16);   // tensor_dim0[15:0]=128 (bits 63:48)
  g1[2] = (int)(Hn << 16);   // tensor_dim1[15:0]=128
  g1[3] = (int)(Wn << 16);   // tile_dim0=128 (bits 127:112)
  g1[4] = (int)Hn;           // tile_dim1=128, tile_dim2=0
  g1[5] = (int)Wn;           // tensor_dim0_stride[31:0]=128
  g1[6] = 0;
  g1[7] = 0;

  i32x4 z4 = {0, 0, 0, 0};   // 2D tensor: groups 2/3 unused
#if TDM_ARGS == 6
  i32x8 z8 = {0, 0, 0, 0, 0, 0, 0, 0};
  __builtin_amdgcn_tensor_load_to_lds(g0, g1, z4, z4, z8, 0);
#else
  __builtin_amdgcn_tensor_load_to_lds(g0, g1, z4, z4, 0);
#endif
}
#endif // HAVE_TDM

__global__ __launch_bounds__(THREADS)
void border_align_tdm_kernel(const float* __restrict__ feature,
                             const float* __restrict__ boxes,
                             float* __restrict__ out) {
  // Two channel planes staged per block: 128 KB of the WGP's 320 KB LDS.
  __shared__ float tile[NCH][Hn * Wn];

  const int bid = blockIdx.x;
  const int kt  = bid & (KT - 1);              // 3 bits
  const int cp  = (bid >> 3) & (Cn / NCH - 1); // 4 bits: channel pair
  const int g   = (bid >> 7) & 3;              // 2 bits
  const int b   = bid >> 9;                    // 1 bit
  const int c0  = cp * NCH;

  // Border g samples channel group g: plane index (b*4 + g)*C + c
  const float* plane0 =
      feature + (size_t)((b * 4 + g) * Cn + c0) * (size_t)(Hn * Wn);

#if HAVE_TDM
  if (threadIdx.x < 32) {                 // wave 0 drives both DMA copies
    tdm_load_plane(plane0, &tile[0][0]);
    tdm_load_plane(plane0 + (size_t)(Hn * Wn), &tile[1][0]);
  }
#else
  for (int i = threadIdx.x; i < (NCH * Hn * Wn) / 4; i += THREADS) {
    ((float4*)&tile[0][0])[i] = ((const float4*)plane0)[i];
  }
#endif

  // --- Overlap with the tensor DMA: prefetch boxes + pre-transform ---------
  const int    k0   = kt * KTILE;
  const float* boxb = boxes + (size_t)b * Kn * 4;

  float4 pb[KPT];  // (ox, oy, ddx, ddy) per owned box
#pragma unroll
  for (int j = 0; j < KPT; ++j) {
    const int k = k0 + (int)threadIdx.x + j * THREADS;
    const float4 bx = *(const float4*)(boxb + (size_t)k * 4);
    const float x1 = bx.x, y1 = bx.y, x2 = bx.z, y2 = bx.w;
    // Border 0: top (y=y1, x sweeps); 1: left (x=x1, y sweeps);
    // Border 2: bottom (y=y2, x sweeps); 3: right (x=x2, y sweeps).
    float4 t;
    if (g == 0)      { t.x = x1; t.y = y1; t.z = x2 - x1; t.w = 0.0f;    }
    else if (g == 1) { t.x = x1; t.y = y1; t.z = 0.0f;    t.w = y2 - y1; }
    else if (g == 2) { t.x = x1; t.y = y2; t.z = x2 - x1; t.w = 0.0f;    }
    else             { t.x = x2; t.y = y1; t.z = 0.0f;    t.w = y2 - y1; }
    pb[j] = t;
  }

#if HAVE_TDM
  if (threadIdx.x < 32) {
    __builtin_amdgcn_s_wait_tensorcnt(0);  // both planes resident in LDS
  }
#endif
  __syncthreads();

  float* outb0 = out + ((size_t)b * Cn + c0) * (size_t)Kn * 4 + g;

#pragma unroll
  for (int j = 0; j < KPT; ++j) {
    const int   k   = k0 + (int)threadIdx.x + j * THREADS;
    const float ox  = pb[j].x, oy = pb[j].y;
    const float ddx = pb[j].z, ddy = pb[j].w;

    float m0 = -__builtin_inff();
    float m1 = -__builtin_inff();
#pragma unroll
    for (int p = 0; p < Pn; ++p) {
      const float t  = (float)p / 10.0f;
      const float xs = ox + t * ddx;
      const float ys = oy + t * ddy;
      const bool valid = (xs > -1.0f) && (xs < (float)Wn) &&
                         (ys > -1.0f) && (ys < (float)Hn);
      const float xc = fminf(fmaxf(xs, 0.0f), (float)(Wn - 1));
      const float yc = fminf(fmaxf(ys, 0.0f), (float)(Hn - 1));
      const int   xi = (int)xc;           // xc >= 0 -> trunc == floor
      const int   yi = (int)yc;
      const float lx = xc - (float)xi;
      const float ly = yc - (float)yi;
      const int   xi1 = (xi + 1 > Wn - 1) ? (Wn - 1) : (xi + 1);
      const int   yi1 = (yi + 1 > Hn - 1) ? (Hn - 1) : (yi + 1);
      const int   i00 = yi  * Wn + xi,  i01 = yi  * Wn + xi1;
      const int   i10 = yi1 * Wn + xi,  i11 = yi1 * Wn + xi1;
      const float w00 = (1.0f - ly) * (1.0f - lx);
      const float w01 = (1.0f - ly) * lx;
      const float w10 = ly * (1.0f - lx);
      const float w11 = ly * lx;

      float v0 = tile[0][i00] * w00 + tile[0][i01] * w01 +
                 tile[0][i10] * w10 + tile[0][i11] * w11;
      float v1 = tile[1][i00] * w00 + tile[1][i01] * w01 +
                 tile[1][i10] * w10 + tile[1][i11] * w11;
      v0 = valid ? v0 : 0.0f;
      v1 = valid ? v1 : 0.0f;
      m0 = fmaxf(m0, v0);
      m1 = fmaxf(m1, v1);
    }
    outb0[(size_t)k * 4]                     = m0;  // out[b][c0  ][k][g]
    outb0[(size_t)k * 4 + (size_t)Kn * 4]    = m1;  // out[b][c0+1][k][g]
  }
}

extern "C" void kernel_launch(void* const* d_in, const int* in_sizes, int n_in,
                              void* d_out, int out_size, void* d_ws,
                              size_t ws_size, hipStream_t stream) {
  (void)in_sizes; (void)n_in; (void)out_size; (void)d_ws; (void)ws_size;
  const float* feature = (const float*)d_in[0];
  const float* boxes   = (const float*)d_in[1];
  // d_in[2] = pool_size (==10) baked in at compile time.
  float* out = (float*)d_out;

  dim3 grid(Bn * 4 * (Cn / NCH) * KT);  // 1024 blocks
  dim3 block(THREADS);                  // 8 wave32 waves
  hipLaunchKernelGGL(border_align_tdm_kernel, grid, block, 0, stream,
                     feature, boxes, out);
}